// Encoder_42502996361299
// MI455X (gfx1250) — compile-verified
//
#include <hip/hip_runtime.h>

typedef __attribute__((ext_vector_type(2))) float v2f;
typedef __attribute__((ext_vector_type(8))) float v8f;

#define N_NODES 100000
#define N_EDGES 3200000
#define D_IN 128
#define D_OUT 128

// ---------------------------------------------------------------------------
// Kernel 1: zero the workspace accumulators (neigh_sum + cnt), float4-wide.
// Total floats = N_NODES*129 = 12,900,000 (divisible by 4).
// ---------------------------------------------------------------------------
__global__ void __launch_bounds__(256) zero_ws_kernel(float4* __restrict__ ws, int n4) {
    int i = blockIdx.x * blockDim.x + threadIdx.x;
    int stride = gridDim.x * blockDim.x;
    float4 z = make_float4(0.f, 0.f, 0.f, 0.f);
    for (; i < n4; i += stride) ws[i] = z;
}

// ---------------------------------------------------------------------------
// Kernel 2: edge scatter. One edge per wave32. Lane l covers feature dims
// [4l, 4l+4). Undirected: accumulate both directions. All feature/accumulator
// traffic is L2-resident (51.2 MB each << 192 MB L2), atomics resolve at L2.
// ---------------------------------------------------------------------------
__device__ __forceinline__ void atomic_add_f32(float* p, float v) {
    __hip_atomic_fetch_add(p, v, __ATOMIC_RELAXED, __HIP_MEMORY_SCOPE_AGENT);
}

__global__ void __launch_bounds__(256) edge_scatter_kernel(
        const float* __restrict__ feat,
        const int*   __restrict__ ei,     // [2, N_EDGES] row-major
        float*       __restrict__ nsum,   // [N_NODES, D_IN]
        float*       __restrict__ cnt) {  // [N_NODES]
    int wave = (blockIdx.x * blockDim.x + threadIdx.x) >> 5;
    int lane = threadIdx.x & 31;
    if (wave >= N_EDGES) return;

    int src = ei[wave];
    int dst = ei[N_EDGES + wave];

    if (lane == 0) {
        atomic_add_f32(&cnt[src], 1.0f);
        atomic_add_f32(&cnt[dst], 1.0f);
    }

    const float4 fd = ((const float4*)(feat + (size_t)dst * D_IN))[lane];
    const float4 fs = ((const float4*)(feat + (size_t)src * D_IN))[lane];

    float* ps = nsum + (size_t)src * D_IN + lane * 4;
    float* pd = nsum + (size_t)dst * D_IN + lane * 4;
    atomic_add_f32(ps + 0, fd.x);
    atomic_add_f32(ps + 1, fd.y);
    atomic_add_f32(ps + 2, fd.z);
    atomic_add_f32(ps + 3, fd.w);
    atomic_add_f32(pd + 0, fs.x);
    atomic_add_f32(pd + 1, fs.y);
    atomic_add_f32(pd + 2, fs.z);
    atomic_add_f32(pd + 3, fs.w);
}

// ---------------------------------------------------------------------------
// Kernel 3: fused mean + GEMM + bias + ReLU with V_WMMA_F32_16X16X4_F32.
// out[16r : 16r+16, 16w : 16w+16] per wave. combined = [feat | nsum*inv],
// equivalent to feat @ W[0:128] + (nsum*inv) @ W[128:256].
//
// A fragment (16x4 f32): lanes 0-15 -> M=lane, K offsets {0,1};
//                        lanes 16-31 -> M=lane-16, K offsets {2,3}.
// B fragment (4x16 f32): mirrored (K in VGPR slot, N=lane&15).
// C/D (16x16 f32, 8 VGPRs): VGPR r -> M = r + 8*(lane>=16), N = lane&15.
// 100000 rows = 6250 blocks * 16 rows exactly -> EXEC all-ones (WMMA req).
// ---------------------------------------------------------------------------
__global__ void __launch_bounds__(256) sage_gemm_kernel(
        const float* __restrict__ feat,
        const float* __restrict__ nsum,
        const float* __restrict__ cnt,
        const float* __restrict__ W,     // [2*D_IN, D_OUT] row-major
        const float* __restrict__ bias,  // [D_OUT]
        float*       __restrict__ out) { // [N_NODES, D_OUT]
    const int wave  = threadIdx.x >> 5;          // 0..7 -> column tile
    const int lane  = threadIdx.x & 31;
    const int row0  = blockIdx.x * 16;
    const int col0  = wave * 16;

    const int m     = lane & 15;                 // A row within tile
    const int khalf = (lane >> 4) * 2;           // K sub-offset: 0 or 2
    const size_t rowA = (size_t)(row0 + m);

    const float inv = 1.0f / fmaxf(cnt[rowA], 1.0f);

    v8f acc = {};

    // --- K = 0..127 : self features @ W_top ---
    const float* arow = feat + rowA * D_IN;
    #pragma unroll
    for (int k0 = 0; k0 < 128; k0 += 4) {
        v2f a = *(const v2f*)(arow + k0 + khalf);
        v2f bf;
        bf.x = W[(size_t)(k0 + khalf + 0) * D_OUT + col0 + m];
        bf.y = W[(size_t)(k0 + khalf + 1) * D_OUT + col0 + m];
        acc = __builtin_amdgcn_wmma_f32_16x16x4_f32(
                false, a, false, bf, (short)0, acc, false, false);
    }

    // --- K = 128..255 : neighbor mean (nsum * inv) @ W_bot ---
    const float* srow = nsum + rowA * D_IN;
    #pragma unroll
    for (int k0 = 0; k0 < 128; k0 += 4) {
        v2f a = *(const v2f*)(srow + k0 + khalf);
        a.x *= inv;
        a.y *= inv;
        v2f bf;
        bf.x = W[(size_t)(128 + k0 + khalf + 0) * D_OUT + col0 + m];
        bf.y = W[(size_t)(128 + k0 + khalf + 1) * D_OUT + col0 + m];
        acc = __builtin_amdgcn_wmma_f32_16x16x4_f32(
                false, a, false, bf, (short)0, acc, false, false);
    }

    // --- bias + ReLU + store ---
    const int n     = lane & 15;
    const int mbase = (lane >> 4) * 8;
    const float bv  = bias[col0 + n];
    #pragma unroll
    for (int r = 0; r < 8; ++r) {
        size_t orow = (size_t)(row0 + mbase + r);
        float v = acc[r] + bv;
        out[orow * D_OUT + col0 + n] = fmaxf(v, 0.0f);
    }
}

// ---------------------------------------------------------------------------
extern "C" void kernel_launch(void* const* d_in, const int* in_sizes, int n_in,
                              void* d_out, int out_size, void* d_ws, size_t ws_size,
                              hipStream_t stream) {
    // setup_inputs order: nodes, features, edge_index, W, b
    const float* feat = (const float*)d_in[1];
    const int*   ei   = (const int*)  d_in[2];
    const float* W    = (const float*)d_in[3];
    const float* bias = (const float*)d_in[4];
    float*       out  = (float*)d_out;

    float* nsum = (float*)d_ws;                       // N_NODES * D_IN floats
    float* cnt  = nsum + (size_t)N_NODES * D_IN;      // N_NODES floats

    // 1) zero accumulators (every call -> deterministic graph replay)
    const int n4 = (N_NODES * D_IN + N_NODES) / 4;    // 3,225,000 float4s
    zero_ws_kernel<<<2048, 256, 0, stream>>>((float4*)d_ws, n4);

    // 2) edge scatter: one edge per wave32, 8 waves per 256-thread block
    const int scatter_blocks = N_EDGES / 8;           // 400,000
    edge_scatter_kernel<<<scatter_blocks, 256, 0, stream>>>(feat, ei, nsum, cnt);

    // 3) fused mean + GEMM(+bias,ReLU): 16 rows/block, 8 col-tiles/block
    const int gemm_blocks = N_NODES / 16;             // 6,250 exactly
    sage_gemm_kernel<<<gemm_blocks, 256, 0, stream>>>(feat, nsum, cnt, W, bias, out);
}